// GCN_3_layer_edge_weight_45311904973170
// MI455X (gfx1250) — compile-verified
//
#include <hip/hip_runtime.h>

// ---------------------------------------------------------------------------
// GCN (3 hidden layers + residual) for MI455X / gfx1250, wave32.
//  * Dense GEMMs: V_WMMA_F32_16X16X4_F32 (full fp32; GEMMs are <5% of work).
//  * Aggregation: CSR (built per call: histogram + LDS scan + reorder into
//    interleaved {src, coef} 8B records), then one-wave-per-node register-
//    accumulated gather: b64 edge-record broadcasts + b64 per-lane feature
//    loads — no f32 atomics in the hot path, all traffic L2-resident.
// ---------------------------------------------------------------------------

typedef __attribute__((ext_vector_type(2))) float v2f;
typedef __attribute__((ext_vector_type(8))) float v8f;

#define IN_F 128
#define H_F  64
#define C_F  32

static inline int ceil_div_h(long a, long b) { return (int)((a + b - 1) / b); }

// ---------------------------- setup kernels --------------------------------

__global__ void fill_zero32_kernel(unsigned int* __restrict__ p, long n) {
    long i = (long)blockIdx.x * blockDim.x + threadIdx.x;
    if (i < n) p[i] = 0u;
}

// weighted in-degree (f32) + in-degree histogram (int) in one pass
__global__ void deg_hist_kernel(const int* __restrict__ dst, const float* __restrict__ ew,
                                float* __restrict__ deg, int* __restrict__ counts, int E) {
    int e = blockIdx.x * blockDim.x + threadIdx.x;
    if (e < E) {
        int d = dst[e];
        atomicAdd(&deg[d], ew[e]);
        atomicAdd(&counts[d], 1);
    }
}

// in-place: deg -> norm = deg>0 ? rsqrt(max(deg,1e-12)) : 0
__global__ void norm_kernel(float* __restrict__ deg, int n) {
    int i = blockIdx.x * blockDim.x + threadIdx.x;
    if (i < n) {
        float d = deg[i];
        deg[i] = (d > 0.0f) ? rsqrtf(fmaxf(d, 1e-12f)) : 0.0f;
    }
}

// --- 3-phase exclusive scan of counts -> rowptr (block size 256) ---

__global__ void scan_block_kernel(const int* __restrict__ counts, int* __restrict__ rowptr,
                                  int* __restrict__ bsums, int n) {
    __shared__ int sdata[256];
    int tid = threadIdx.x;
    int i = blockIdx.x * 256 + tid;
    int v = (i < n) ? counts[i] : 0;
    sdata[tid] = v;
    __syncthreads();
#pragma unroll
    for (int off = 1; off < 256; off <<= 1) {
        int t = (tid >= off) ? sdata[tid - off] : 0;
        __syncthreads();
        sdata[tid] += t;
        __syncthreads();
    }
    if (i < n) rowptr[i] = sdata[tid] - v;           // exclusive within block
    if (tid == 255) bsums[blockIdx.x] = sdata[255];  // block total
}

__global__ void scan_sums_kernel(int* __restrict__ bsums, int nb) {
    if (blockIdx.x == 0 && threadIdx.x == 0) {
        int acc = 0;
        for (int i = 0; i < nb; ++i) { int t = bsums[i]; bsums[i] = acc; acc += t; }
    }
}

__global__ void scan_add_kernel(int* __restrict__ rowptr, const int* __restrict__ bsums,
                                int n, int E) {
    int i = blockIdx.x * blockDim.x + threadIdx.x;
    if (i < n) rowptr[i] += bsums[i >> 8];
    if (i == 0) rowptr[n] = E;
}

// per-edge coefficient + bucketed reorder into CSR order, interleaved records
__global__ void build_edges_kernel(const int* __restrict__ src, const int* __restrict__ dst,
                                   const float* __restrict__ ew, const float* __restrict__ norm,
                                   const int* __restrict__ rowptr, int* __restrict__ cursor,
                                   int2* __restrict__ eRec, int E) {
    int e = blockIdx.x * blockDim.x + threadIdx.x;
    if (e < E) {
        int s = src[e], d = dst[e];
        float c = ew[e] * norm[s] * norm[d];
        int pos = rowptr[d] + atomicAdd(&cursor[d], 1);
        eRec[pos] = make_int2(s, __float_as_int(c));   // one b64 store
    }
}

// ------------------------------- WMMA GEMM ---------------------------------
// Y[n, FOUT] = X[n, FIN] @ W[FIN, FOUT]
template <int FIN, int FOUT>
__global__ void gemm_wmma_kernel(const float* __restrict__ X, const float* __restrict__ W,
                                 float* __restrict__ Y, int nrows) {
    __shared__ float tx[16 * FIN];
    const int nthreads = (FOUT / 16) * 32;
    const int m0  = blockIdx.x * 16;
    const int tid = threadIdx.x;

    for (int i = tid; i < 16 * FIN; i += nthreads) {
        int r = i / FIN, c = i % FIN;
        int row = m0 + r;
        if (row >= nrows) row = nrows - 1;
        tx[i] = X[(long)row * FIN + c];
    }
    __syncthreads();

    const int wave  = tid >> 5;
    const int lane  = tid & 31;
    const int lrow  = lane & 15;
    const int khalf = (lane >> 4) * 2;
    const int n0    = wave * 16;

    v8f acc = {};
#pragma unroll
    for (int k0 = 0; k0 < FIN; k0 += 4) {
        v2f a = *(const v2f*)&tx[lrow * FIN + k0 + khalf];
        v2f b;
        b.x = W[(long)(k0 + khalf + 0) * FOUT + n0 + lrow];
        b.y = W[(long)(k0 + khalf + 1) * FOUT + n0 + lrow];
        acc = __builtin_amdgcn_wmma_f32_16x16x4_f32(
            false, a, false, b, (short)0, acc, false, false);
    }

    const int mbase = m0 + ((lane >> 4) * 8);
    const int col   = n0 + lrow;
#pragma unroll
    for (int r = 0; r < 8; ++r) {
        int row = mbase + r;
        if (row < nrows) Y[(long)row * FOUT + col] = acc[r];
    }
}

// ------------------------- CSR gather-aggregate ----------------------------
// y[node,:] = act( sum_{e in in(node)} h[src_e,:]*coef_e + bias
//                  [+ res + bias2 if FINAL] )
// One wave per destination node; lane covers features [VPL*lane, VPL*lane+VPL).
template <int F, bool RELU, bool FINAL>
__global__ void gather_agg_kernel(const float* __restrict__ h,
                                  const int* __restrict__ rowptr,
                                  const int2* __restrict__ eRec,
                                  const float* __restrict__ bias,
                                  const float* __restrict__ res,
                                  const float* __restrict__ bias2,
                                  float* __restrict__ y, int n) {
    const int wave = threadIdx.x >> 5;
    const int lane = threadIdx.x & 31;
    const int node = blockIdx.x * (blockDim.x >> 5) + wave;
    if (node >= n) return;

    const int s0 = rowptr[node];
    const int s1 = rowptr[node + 1];

    constexpr int VPL = F / 32;          // floats per lane (2 for F=64, 1 for F=32)
    const int fbase = lane * VPL;

    float acc[VPL];
#pragma unroll
    for (int j = 0; j < VPL; ++j) acc[j] = 0.0f;

    for (int i = s0; i < s1; ++i) {
        int2  r = eRec[i];               // wave-uniform b64 broadcast
        float c = __int_as_float(r.y);
        const float* hr = h + (long)r.x * F + fbase;
        if constexpr (VPL == 2) {
            float2 v = *(const float2*)hr;   // one b64 per lane
            acc[0] += v.x * c;
            acc[1] += v.y * c;
        } else {
            acc[0] += hr[0] * c;
        }
    }

    const long base = (long)node * F + fbase;
#pragma unroll
    for (int j = 0; j < VPL; ++j) {
        float v = acc[j] + bias[fbase + j];
        if (FINAL) v += res[base + j] + bias2[fbase + j];
        if (RELU) v = fmaxf(v, 0.0f);
        acc[j] = v;
    }
    if constexpr (VPL == 2) {
        *(float2*)(y + base) = make_float2(acc[0], acc[1]);   // one b64 store
    } else {
        y[base] = acc[0];
    }
}

// ------------------------------- launcher ----------------------------------

extern "C" void kernel_launch(void* const* d_in, const int* in_sizes, int n_in,
                              void* d_out, int out_size, void* d_ws, size_t ws_size,
                              hipStream_t stream) {
    const float* x    = (const float*)d_in[0];
    const int*   src  = (const int*)  d_in[1];
    const int*   dst  = (const int*)  d_in[2];
    const float* ew   = (const float*)d_in[3];
    const float* W1   = (const float*)d_in[4];
    const float* b1   = (const float*)d_in[5];
    const float* W2   = (const float*)d_in[6];
    const float* b2   = (const float*)d_in[7];
    const float* W3   = (const float*)d_in[8];
    const float* b3   = (const float*)d_in[9];
    const float* W4   = (const float*)d_in[10];
    const float* b4   = (const float*)d_in[11];
    const float* Wres = (const float*)d_in[12];
    const float* bres = (const float*)d_in[13];
    float* out = (float*)d_out;

    const int N  = in_sizes[0] / IN_F;
    const int E  = in_sizes[1];
    const int nb = ceil_div_h(N, 256);  // scan blocks

    // workspace layout (32-bit words); deg/counts/cursor contiguous for one fill
    unsigned int* ws = (unsigned int*)d_ws;
    float* deg    = (float*)(ws);                       // N   (becomes norm)
    int*   counts = (int*)  (ws + N);                   // N
    int*   cursor = (int*)  (ws + 2L * N);              // N
    int*   rowptr = (int*)  (ws + 3L * N);              // N + 1
    int*   bsums  = (int*)  (ws + 4L * N + 1);          // nb
    long   recOff = (4L * N + 1 + nb + 1) & ~1L;        // 8B-align for int2
    int2*  eRec   = (int2*) (ws + recOff);              // E records (8B each)
    float* h      = (float*)(ws + recOff + 2L * E);     // N * 64
    float* xbuf   = h    + (size_t)N * H_F;             // N * 64
    float* res    = xbuf + (size_t)N * H_F;             // N * 32

    const int T  = 256;
    const int mt = ceil_div_h(N, 16);
    const int gblocks = ceil_div_h(N, T / 32);          // gather: 1 wave / node

    // --- CSR + normalization build ---
    fill_zero32_kernel<<<ceil_div_h(3L * N, T), T, 0, stream>>>(ws, 3L * N);
    deg_hist_kernel<<<ceil_div_h(E, T), T, 0, stream>>>(dst, ew, deg, counts, E);
    norm_kernel<<<ceil_div_h(N, T), T, 0, stream>>>(deg, N);
    scan_block_kernel<<<nb, 256, 0, stream>>>(counts, rowptr, bsums, N);
    scan_sums_kernel<<<1, 32, 0, stream>>>(bsums, nb);
    scan_add_kernel<<<ceil_div_h(N, T), T, 0, stream>>>(rowptr, bsums, N, E);
    build_edges_kernel<<<ceil_div_h(E, T), T, 0, stream>>>(src, dst, ew, deg, rowptr, cursor,
                                                           eRec, E);

    // --- residual: res = inputs @ Wres (bres folded into final gather) ---
    gemm_wmma_kernel<IN_F, C_F><<<mt, (C_F / 16) * 32, 0, stream>>>(x, Wres, res, N);

    // --- layer 1: 128 -> 64, relu ---
    gemm_wmma_kernel<IN_F, H_F><<<mt, (H_F / 16) * 32, 0, stream>>>(x, W1, h, N);
    gather_agg_kernel<H_F, true, false><<<gblocks, T, 0, stream>>>(h, rowptr, eRec,
                                                                   b1, nullptr, nullptr, xbuf, N);
    // --- layer 2: 64 -> 64, relu ---
    gemm_wmma_kernel<H_F, H_F><<<mt, (H_F / 16) * 32, 0, stream>>>(xbuf, W2, h, N);
    gather_agg_kernel<H_F, true, false><<<gblocks, T, 0, stream>>>(h, rowptr, eRec,
                                                                   b2, nullptr, nullptr, xbuf, N);
    // --- layer 3: 64 -> 64, relu ---
    gemm_wmma_kernel<H_F, H_F><<<mt, (H_F / 16) * 32, 0, stream>>>(xbuf, W3, h, N);
    gather_agg_kernel<H_F, true, false><<<gblocks, T, 0, stream>>>(h, rowptr, eRec,
                                                                   b3, nullptr, nullptr, xbuf, N);
    // --- layer 4: 64 -> 32, + b4 + res + bres, no relu, straight to d_out ---
    gemm_wmma_kernel<H_F, C_F><<<mt, (C_F / 16) * 32, 0, stream>>>(xbuf, W4, h, N);
    gather_agg_kernel<C_F, false, true><<<gblocks, T, 0, stream>>>(h, rowptr, eRec,
                                                                   b4, res, bres, out, N);
}